// SAGPool_58334245814482
// MI455X (gfx1250) — compile-verified
//
#include <hip/hip_runtime.h>
#include <math.h>

#define NN 50000
#define NE 800000
#define HID 128
#define NG 64
#define HDIM 384          // 3*HID (concat width)
#define PDIM 768          // pooled width (avg|max)

typedef __attribute__((ext_vector_type(2))) float v2f;
typedef __attribute__((ext_vector_type(8))) float v8f;

// ---------------------------------------------------------------------------
// Y[n_rows x 128] = X[n_rows x K] @ W[K x 128], exact f32 via V_WMMA_F32_16X16X4_F32
// One wave handles a 16-row tile across all 8 column tiles.
// A layout (16x4 f32): lanes 0-15 M=0..15 hold K={k,k+1}; lanes 16-31 hold K={k+2,k+3}
// B layout (4x16 f32): VGPR0 = row k (lanes 0-15) / row k+2 (lanes 16-31); VGPR1 = k+1 / k+3
// C/D layout: VGPR v -> M = v + 8*half, N = lane&15 within the tile
// ---------------------------------------------------------------------------
__global__ __launch_bounds__(128)
void gemm128_wmma(const float* __restrict__ X, const float* __restrict__ W,
                  float* __restrict__ Y, int n_rows, int K) {
  const int wave = threadIdx.x >> 5;
  const int lane = threadIdx.x & 31;
  const int n_tiles = n_rows >> 4;
  const int row_tile = blockIdx.x * 4 + wave;
  if (row_tile >= n_tiles) return;            // wave-uniform: EXEC stays all-ones
  const int half = lane >> 4;                 // 0 or 1
  const int m    = lane & 15;
  const long row = (long)row_tile * 16 + m;

  v8f acc[8];
#pragma unroll
  for (int t = 0; t < 8; ++t) acc[t] = 0.f;

  const float* xrow = X + row * (long)K;
  for (int k = 0; k < K; k += 4) {
    v2f a;
    a.x = xrow[k + 2 * half + 0];
    a.y = xrow[k + 2 * half + 1];
    const float* wp = W + (long)(k + 2 * half) * HID + m;
#pragma unroll
    for (int t = 0; t < 8; ++t) {
      v2f b;
      b.x = wp[t * 16];
      b.y = wp[t * 16 + HID];
      acc[t] = __builtin_amdgcn_wmma_f32_16x16x4_f32(
          false, a, false, b, (short)0, acc[t], false, false);
    }
  }

#pragma unroll
  for (int t = 0; t < 8; ++t)
#pragma unroll
    for (int v = 0; v < 8; ++v)
      Y[((long)row_tile * 16 + v + half * 8) * HID + t * 16 + m] = acc[t][v];
}

// Y[n] = b[n % 128]  (seed accumulation buffer with the bias)
__global__ void init_bias_k(float* __restrict__ Y, const float* __restrict__ b, int n) {
  int i = blockIdx.x * blockDim.x + threadIdx.x;
  if (i < n) Y[i] = b[i & (HID - 1)];
}

// Per edge e: Y[dst] += w_e * S[src]  (wave per edge, lane covers 4 of 128 dims)
__global__ __launch_bounds__(256)
void scatter_edges_k(const float* __restrict__ S, const int* __restrict__ ei,
                     const float* __restrict__ ew, float* __restrict__ Y) {
  int e    = (blockIdx.x * blockDim.x + threadIdx.x) >> 5;
  int lane = threadIdx.x & 31;
  if (e >= NE) return;
  int   src = ei[e];
  int   dst = ei[NE + e];
  float w   = ew[e];
  const float4 v = *(const float4*)(S + (long)src * HID + lane * 4);
  float* o = Y + (long)dst * HID + lane * 4;
  atomicAdd(o + 0, w * v.x);
  atomicAdd(o + 1, w * v.y);
  atomicAdd(o + 2, w * v.z);
  atomicAdd(o + 3, w * v.w);
}

__global__ void relu_k(float* __restrict__ Y, int n) {
  int i = blockIdx.x * blockDim.x + threadIdx.x;
  if (i < n) Y[i] = fmaxf(Y[i], 0.f);
}

// s[n] = h[n,:] . wa  with h = [g1|g2|g3]; wave per node, shuffle reduction
__global__ __launch_bounds__(256)
void att_dot_k(const float* __restrict__ g1, const float* __restrict__ g2,
               const float* __restrict__ g3, const float* __restrict__ wa,
               float* __restrict__ s) {
  int n    = (blockIdx.x * blockDim.x + threadIdx.x) >> 5;
  int lane = threadIdx.x & 31;
  if (n >= NN) return;
  const float* r1 = g1 + (long)n * HID;
  const float* r2 = g2 + (long)n * HID;
  const float* r3 = g3 + (long)n * HID;
  float sum = 0.f;
  for (int j = lane; j < HID; j += 32)
    sum += r1[j] * wa[j] + r2[j] * wa[HID + j] + r3[j] * wa[2 * HID + j];
  for (int off = 16; off > 0; off >>= 1) sum += __shfl_down(sum, off, 32);
  if (lane == 0) s[n] = sum;
}

__global__ void init_scalar_k(float* __restrict__ y, const float* __restrict__ v, int n) {
  int i = blockIdx.x * blockDim.x + threadIdx.x;
  if (i < n) y[i] = v[0];
}

__global__ void scatter_scalar_k(const float* __restrict__ s, const int* __restrict__ ei,
                                 const float* __restrict__ ew, float* __restrict__ score) {
  int e = blockIdx.x * blockDim.x + threadIdx.x;
  if (e < NE) atomicAdd(&score[ei[NE + e]], ew[e] * s[ei[e]]);
}

__global__ void tanh_k(float* __restrict__ y, int n) {
  int i = blockIdx.x * blockDim.x + threadIdx.x;
  if (i < n) y[i] = tanhf(y[i]);
}

// One block per graph: graph_indicator is sorted -> binary search segment bounds,
// then avg/max reduce pool = h*score over the node range. pooled[g] = [avg(384)|max(384)]
__global__ __launch_bounds__(128)
void pool_reduce_k(const float* __restrict__ g1, const float* __restrict__ g2,
                   const float* __restrict__ g3, const float* __restrict__ score,
                   const int* __restrict__ gid, float* __restrict__ pooled) {
  int g = blockIdx.x;
  int lo = 0, hi = NN;
  while (lo < hi) { int mid = (lo + hi) >> 1; if (gid[mid] < g) lo = mid + 1; else hi = mid; }
  int start = lo;
  lo = start; hi = NN;
  while (lo < hi) { int mid = (lo + hi) >> 1; if (gid[mid] < g + 1) lo = mid + 1; else hi = mid; }
  int end = lo;
  float inv = 1.f / fmaxf((float)(end - start), 1.f);
  for (int d = threadIdx.x; d < HDIM; d += blockDim.x) {
    const float* buf = (d < HID) ? g1 : (d < 2 * HID) ? g2 : g3;
    int dd = d & (HID - 1);
    float sum = 0.f, mx = -INFINITY;
    for (int n = start; n < end; ++n) {
      float v = buf[(long)n * HID + dd] * score[n];
      sum += v;
      mx = fmaxf(mx, v);
    }
    pooled[g * PDIM + d]        = sum * inv;
    pooled[g * PDIM + HDIM + d] = mx;
  }
}

__global__ void bias_relu_k(const float* __restrict__ Yin, const float* __restrict__ bf,
                            float* __restrict__ out, int n) {
  int i = blockIdx.x * blockDim.x + threadIdx.x;
  if (i < n) out[i] = fmaxf(Yin[i] + bf[i & (HID - 1)], 0.f);
}

// ---------------------------------------------------------------------------
extern "C" void kernel_launch(void* const* d_in, const int* in_sizes, int n_in,
                              void* d_out, int out_size, void* d_ws, size_t ws_size,
                              hipStream_t stream) {
  const int*   ei  = (const int*)d_in[0];     // edge_index [2, E]
  const float* ew  = (const float*)d_in[1];   // edge_weight [E]
  const float* xf  = (const float*)d_in[2];   // input_feature [N, 128]
  const int*   gid = (const int*)d_in[3];     // graph_indicator [N] (sorted)
  const float* W1  = (const float*)d_in[4];
  const float* b1  = (const float*)d_in[5];
  const float* W2  = (const float*)d_in[6];
  const float* b2  = (const float*)d_in[7];
  const float* W3  = (const float*)d_in[8];
  const float* b3  = (const float*)d_in[9];
  const float* wa  = (const float*)d_in[10];  // [384,1]
  const float* ba  = (const float*)d_in[11];  // [1]
  const float* Wf  = (const float*)d_in[12];  // [768,128]
  const float* bf  = (const float*)d_in[13];  // [128]
  float* out = (float*)d_out;                 // [64,128]

  // workspace carve-up (256B aligned)
  char*  ws  = (char*)d_ws;
  size_t off = 0;
  auto carve = [&](size_t bytes) {
    void* p = ws + off;
    off += (bytes + 255) & ~(size_t)255;
    return p;
  };
  const size_t featB = (size_t)NN * HID * sizeof(float);
  float* support = (float*)carve(featB);
  float* g1      = (float*)carve(featB);
  float* g2      = (float*)carve(featB);
  float* g3      = (float*)carve(featB);
  float* sdot    = (float*)carve((size_t)NN * sizeof(float));
  float* score   = (float*)carve((size_t)NN * sizeof(float));
  float* pooled  = (float*)carve((size_t)NG * PDIM * sizeof(float));
  float* ytmp    = (float*)carve((size_t)NG * HID * sizeof(float));
  (void)ws_size; (void)n_in; (void)in_sizes; (void)out_size;

  const int feat_n     = NN * HID;
  const int feat_blks  = (feat_n + 255) / 256;
  const int gemm_blks  = ((NN / 16) + 3) / 4;     // 3125 tiles / 4 waves-per-block
  const int edge_wblks = NE / 8;                  // wave-per-edge, 8 waves/block
  const int node_wblks = (NN + 7) / 8;

  const float* layer_in[3] = { xf, g1, g2 };
  const float* layer_W[3]  = { W1, W2, W3 };
  const float* layer_b[3]  = { b1, b2, b3 };
  float*       layer_out[3]= { g1, g2, g3 };

  for (int l = 0; l < 3; ++l) {
    gemm128_wmma<<<gemm_blks, 128, 0, stream>>>(layer_in[l], layer_W[l], support, NN, HID);
    init_bias_k<<<feat_blks, 256, 0, stream>>>(layer_out[l], layer_b[l], feat_n);
    scatter_edges_k<<<edge_wblks, 256, 0, stream>>>(support, ei, ew, layer_out[l]);
    relu_k<<<feat_blks, 256, 0, stream>>>(layer_out[l], feat_n);
  }

  // attention score: s = h @ wa ; score = tanh(ba + segment_sum(w_e * s[src] -> dst))
  att_dot_k<<<node_wblks, 256, 0, stream>>>(g1, g2, g3, wa, sdot);
  init_scalar_k<<<(NN + 255) / 256, 256, 0, stream>>>(score, ba, NN);
  scatter_scalar_k<<<(NE + 255) / 256, 256, 0, stream>>>(sdot, ei, ew, score);
  tanh_k<<<(NN + 255) / 256, 256, 0, stream>>>(score, NN);

  // per-graph avg/max pooling of h*score
  pool_reduce_k<<<NG, 128, 0, stream>>>(g1, g2, g3, score, gid, pooled);

  // final: relu(pooled @ Wf + bf) — reuse the WMMA GEMM (64x768 @ 768x128)
  gemm128_wmma<<<1, 128, 0, stream>>>(pooled, Wf, ytmp, NG, PDIM);
  bias_relu_k<<<(NG * HID + 255) / 256, 256, 0, stream>>>(ytmp, bf, out, NG * HID);
}